// SelfAttention_16535624090167
// MI455X (gfx1250) — compile-verified
//
#include <hip/hip_runtime.h>
#include <math.h>

// Fused "linear attention" kernel for MI455X (gfx1250).
// No matmul exists in this op (per-column dot-reductions + elementwise
// softmax), so WMMA is inapplicable; the CDNA5 features used are the 320KB
// WGP LDS and the async global->LDS copy path (ASYNCcnt).
//
// Block owns (batch b, 16 columns e0..e0+15) x all S=4096 rows.
// LDS tile: 16 cols * 4096 rows * 4B = 256KB, staged via
// global_load_async_to_lds_b128; 64B contiguous per row for DRAM efficiency.

#define TPB   1024
#define EC    16
#define S_DIM 4096
#define E_DIM 1024
#define NF    (S_DIM * EC / 4)   // 16384 float4 chunks per tile

__device__ __forceinline__ float gelu1(float x) {
  // exact gelu (erf-based) + 1 ; always > 0.83 (so strictly positive)
  return 0.5f * x * (1.0f + erff(x * 0.70710678118654752f)) + 1.0f;
}

// Deterministic block-wide reduction. Thread tid holds partials for columns
// (tid&3)*4 .. +3. Returns the column totals (same mapping) to every thread.
// Fixed tree order => bitwise deterministic across runs.
__device__ __forceinline__ float4 block_reduce4(float* rws, int tid, float4 v,
                                                bool is_max) {
  rws[tid * 4 + 0] = v.x; rws[tid * 4 + 1] = v.y;
  rws[tid * 4 + 2] = v.z; rws[tid * 4 + 3] = v.w;
  __syncthreads();
  for (int off = (TPB / 4) / 2; off >= 1; off >>= 1) {  // 256 logical rows/group
    if ((tid >> 2) < off) {
      int s = tid * 4, o = (tid + off * 4) * 4;
      if (is_max) {
        rws[s + 0] = fmaxf(rws[s + 0], rws[o + 0]);
        rws[s + 1] = fmaxf(rws[s + 1], rws[o + 1]);
        rws[s + 2] = fmaxf(rws[s + 2], rws[o + 2]);
        rws[s + 3] = fmaxf(rws[s + 3], rws[o + 3]);
      } else {
        rws[s + 0] += rws[o + 0];
        rws[s + 1] += rws[o + 1];
        rws[s + 2] += rws[o + 2];
        rws[s + 3] += rws[o + 3];
      }
    }
    __syncthreads();
  }
  int p = (tid & 3) * 4;
  float4 r = make_float4(rws[p + 0], rws[p + 1], rws[p + 2], rws[p + 3]);
  __syncthreads();   // protect rws before reuse
  return r;
}

extern "C" __global__ __launch_bounds__(TPB, 1)
void SelfAttention_16535624090167_kernel(const float* __restrict__ query,
                                         const float* __restrict__ keys,
                                         const float* __restrict__ values,
                                         float* __restrict__ out) {
  __shared__ float4 qs[NF];        // 256KB column tile (CDNA5 big-LDS)
  __shared__ float  rws[TPB * 4];  // 16KB reduction scratch

  const int tid = threadIdx.x;
  const int bid = blockIdx.x;
  const int b   = bid >> 6;              // / (E_DIM/EC = 64)
  const int e0  = (bid & 63) * EC;
  const size_t base = (size_t)b * S_DIM * E_DIM + (size_t)e0;
  const float* qt = query  + base;
  const float* kt = keys   + base;
  const float* vt = values + base;
  float*       ot = out    + base;

  // ---- Phase 0: async global -> LDS copy of raw query tile (ASYNCcnt) ----
  {
    unsigned long long gbase = (unsigned long long)(const void*)qt;
    // flat addr of LDS object: low 32 bits == workgroup-relative LDS offset
    unsigned int lds0 = (unsigned int)(unsigned long long)(const void*)&qs[0];
    for (int f = tid; f < NF; f += TPB) {   // 16 b128 chunks issued per lane
      unsigned int goff = (unsigned int)((f >> 2) * (E_DIM * 4) + (f & 3) * 16);
      unsigned int loff = lds0 + (unsigned int)(f * 16);
      asm volatile("global_load_async_to_lds_b128 %0, %1, %2"
                   :: "v"(loff), "v"(goff), "s"(gbase)
                   : "memory");
    }
    asm volatile("s_wait_asynccnt 0" ::: "memory");
  }
  __syncthreads();

  // ---- Phase 1: gelu(Q)+1 in place; stream K; per-column c and max(q) ----
  float4 accC = make_float4(0.f, 0.f, 0.f, 0.f);
  float4 accM = make_float4(0.f, 0.f, 0.f, 0.f);   // q > 0, so 0 is identity
  for (int f = tid; f < NF; f += TPB) {
    float4 q = qs[f];
    q.x = gelu1(q.x); q.y = gelu1(q.y); q.z = gelu1(q.z); q.w = gelu1(q.w);
    qs[f] = q;
    const float4 kv =
        *(const float4*)(kt + (size_t)(f >> 2) * E_DIM + (size_t)((f & 3) * 4));
    float4 kk;
    kk.x = gelu1(kv.x); kk.y = gelu1(kv.y); kk.z = gelu1(kv.z); kk.w = gelu1(kv.w);
    accC.x += q.x * kk.x; accC.y += q.y * kk.y;
    accC.z += q.z * kk.z; accC.w += q.w * kk.w;
    accM.x = fmaxf(accM.x, q.x); accM.y = fmaxf(accM.y, q.y);
    accM.z = fmaxf(accM.z, q.z); accM.w = fmaxf(accM.w, q.w);
  }
  float4 cv = block_reduce4(rws, tid, accC, false);  // c[e] (always > 0)
  float4 mq = block_reduce4(rws, tid, accM, true);   // max_s q[s,e]

  const float inv_scale = 0.03125f;                  // 1/sqrt(1024)
  float4 a = make_float4(cv.x * inv_scale, cv.y * inv_scale,
                         cv.z * inv_scale, cv.w * inv_scale);
  float4 m = make_float4(a.x * mq.x, a.y * mq.y, a.z * mq.z, a.w * mq.w);

  // ---- Phase 2: exp(a*q - m) in place; column sums; prefetch V stream ----
  float4 accS = make_float4(0.f, 0.f, 0.f, 0.f);
  for (int f = tid; f < NF; f += TPB) {
    if ((f & 3) == 0)   // one prefetch per 64B row segment
      __builtin_prefetch(vt + (size_t)(f >> 2) * E_DIM, 0, 0);
    float4 q = qs[f];
    float4 ex;
    ex.x = __expf(a.x * q.x - m.x);
    ex.y = __expf(a.y * q.y - m.y);
    ex.z = __expf(a.z * q.z - m.z);
    ex.w = __expf(a.w * q.w - m.w);
    qs[f] = ex;
    accS.x += ex.x; accS.y += ex.y; accS.z += ex.z; accS.w += ex.w;
  }
  float4 sv = block_reduce4(rws, tid, accS, false);
  float4 r = make_float4(1.0f / sv.x, 1.0f / sv.y, 1.0f / sv.z, 1.0f / sv.w);

  // ---- Phase 3: out = softmax * values ----
  for (int f = tid; f < NF; f += TPB) {
    const size_t go = (size_t)(f >> 2) * E_DIM + (size_t)((f & 3) * 4);
    float4 ex = qs[f];
    float4 v  = *(const float4*)(vt + go);
    float4 o;
    o.x = ex.x * r.x * v.x;
    o.y = ex.y * r.y * v.y;
    o.z = ex.z * r.z * v.z;
    o.w = ex.w * r.w * v.w;
    *(float4*)(ot + go) = o;
  }
}

extern "C" void kernel_launch(void* const* d_in, const int* in_sizes, int n_in,
                              void* d_out, int out_size, void* d_ws, size_t ws_size,
                              hipStream_t stream) {
  const float* q = (const float*)d_in[0];
  const float* k = (const float*)d_in[1];
  const float* v = (const float*)d_in[2];
  float* o = (float*)d_out;
  const int Btot = in_sizes[0] / (S_DIM * E_DIM);   // = 8
  dim3 grid((unsigned)(Btot * (E_DIM / EC)));       // 512 blocks
  dim3 block(TPB);
  SelfAttention_16535624090167_kernel<<<grid, block, 0, stream>>>(q, k, v, o);
}